// SmoothAP_26680336843537
// MI455X (gfx1250) — compile-verified
//
#include <hip/hip_runtime.h>

// SmoothAP for N rows of M=2048.
// Stage 1: per (n, 16-row i-tile) wave computes rank sums with
//          V_WMMA_F32_16X16X4_F32 accumulating  D[i][0]=sum_j sigma,
//          D[i][1]=sum_j sigma*pos[j]  over 512 K=4 steps.
//          Sigmoid via V_TANH_F32 (1 trans op) when available, else exp2+rcp.
// Stage 2: per-n deterministic tree reduction -> per_row[n].
// Stage 3: single-wave reduction -> out = 1 - mean(per_row).

typedef __attribute__((ext_vector_type(2))) float v2f;
typedef __attribute__((ext_vector_type(8))) float v8f;

#define M_DIM 2048
#define TILE_I 16

#if __has_builtin(__builtin_amdgcn_tanhf)
#define USE_TANH 1
#else
#define USE_TANH 0
#endif

// sigmoid((simj - simi)*100) with the reference's clip(x/T, -50, 50).
// Tanh path:  h = clamp((simj - simi)*50, -25, 25);  s = 0.5*tanh(h) + 0.5
// Exp path:   q = clamp((simi - simj)*100*log2e, +-50*log2e);  s = 1/(1+2^q)
__device__ __forceinline__ float sig_fast(float simj, float pre_i) {
#if USE_TANH
    float h = fmaf(simj, 50.0f, pre_i);          // pre_i = -simi*50
    h = fminf(fmaxf(h, -25.0f), 25.0f);          // v_med3
    return fmaf(__builtin_amdgcn_tanhf(h), 0.5f, 0.5f);
#else
    const float C  = 144.269504089f;             // 100 * log2(e)
    const float CL = 72.1347520444f;             // 50 * log2(e)
    float q = fmaf(simj, -C, pre_i);             // pre_i = simi*C
    q = fminf(fmaxf(q, -CL), CL);
    float e = __builtin_amdgcn_exp2f(q);
    return __builtin_amdgcn_rcpf(e + 1.0f);
#endif
}

__global__ __launch_bounds__(32) void smoothap_rank_kernel(
    const float* __restrict__ sim, const float* __restrict__ targ,
    float* __restrict__ contrib) {
    __shared__ float sSim[M_DIM];
    __shared__ float sPos[M_DIM];
    __shared__ float sD[TILE_I * 2];

    const int lane = threadIdx.x;            // 0..31, one wave per block
    const int tilesPerRow = M_DIM / TILE_I;  // 128
    const int n  = blockIdx.x / tilesPerRow;
    const int i0 = (blockIdx.x % tilesPerRow) * TILE_I;

    const float* simRow = sim  + (size_t)n * M_DIM;
    const float* trgRow = targ + (size_t)n * M_DIM;

    // Stage sim row and pos-mask row into LDS (uniform trip count: EXEC stays full).
    {
        const float4* s4 = (const float4*)simRow;
        const float4* t4 = (const float4*)trgRow;
        float4* ls4 = (float4*)sSim;
        float4* lp4 = (float4*)sPos;
        for (int t = lane; t < M_DIM / 4; t += 32) {
            float4 sv = s4[t];
            ls4[t] = sv;
            float4 tv = t4[t];
            float4 pv;
            pv.x = (tv.x > 0.5f) ? 1.0f : 0.0f;
            pv.y = (tv.y > 0.5f) ? 1.0f : 0.0f;
            pv.z = (tv.z > 0.5f) ? 1.0f : 0.0f;
            pv.w = (tv.w > 0.5f) ? 1.0f : 0.0f;
            lp4[t] = pv;
        }
    }
    __syncthreads();

    // A layout (16x4 f32): row = lane&15 for both halves; K = {0,1} (lanes 0-15)
    // or {2,3} (lanes 16-31).  B layout (4x16 f32): column = lane&15, same K split.
    const int   col  = lane & 15;
    const int   half = lane >> 4;
    const float simi = sSim[i0 + col];
#if USE_TANH
    const float pre_i = simi * -50.0f;
#else
    const float pre_i = simi * 144.269504089f;
#endif
    const int   ig   = i0 + col;                       // this lane's global i
    const float isc0 = (col == 0) ? 1.0f : 0.0f;       // B column 0 -> ones
    const float isc1 = (col == 1) ? 1.0f : 0.0f;       // B column 1 -> pos[j]

    v8f c = {0.f, 0.f, 0.f, 0.f, 0.f, 0.f, 0.f, 0.f};

    auto step = [&](int j0, bool dodiag) {
        const int jj = j0 + half * 2;                  // this lane's first j
        v2f sj = *(const v2f*)(sSim + jj);             // ds_load broadcast
        v2f pj = *(const v2f*)(sPos + jj);
        float sx = sig_fast(sj.x, pre_i);
        float sy = sig_fast(sj.y, pre_i);
        if (dodiag) {                                  // compile-time constant
            sx = (jj     == ig) ? 0.0f : sx;           // v_cndmask, no EXEC change
            sy = (jj + 1 == ig) ? 0.0f : sy;
        }
        v2f a; a.x = sx; a.y = sy;
        v2f b;
        b.x = fmaf(isc1, pj.x, isc0);
        b.y = fmaf(isc1, pj.y, isc0);
        c = __builtin_amdgcn_wmma_f32_16x16x4_f32(false, a, false, b,
                                                  (short)0, c, false, false);
    };

    // Diagonal only intersects the j-chunk [i0, i0+16): mask only there.
    #pragma unroll 4
    for (int j0 = 0; j0 < i0; j0 += 4) step(j0, false);
    #pragma unroll
    for (int j0 = i0; j0 < i0 + TILE_I; j0 += 4) step(j0, true);
    #pragma unroll 4
    for (int j0 = i0 + TILE_I; j0 < M_DIM; j0 += 4) step(j0, false);

    // D layout: lane c (c<16) vgpr r = D[r][c]; lane 16+c vgpr r = D[8+r][c].
    // Only columns 0 (sum_all) and 1 (sum_pos) are meaningful.
    if (col < 2) {
        #pragma unroll
        for (int r = 0; r < 8; ++r)
            sD[(half * 8 + r) * 2 + col] = c[r];
    }
    __syncthreads();

    if (lane < TILE_I) {
        float allr = sD[lane * 2 + 0] + 1.0f;          // sim_all_rk
        float posr = sD[lane * 2 + 1] + 1.0f;          // sim_pos_rk
        float pi   = sPos[i0 + lane];
        float ti   = trgRow[i0 + lane];
        contrib[(size_t)n * M_DIM + i0 + lane] = (posr * pi * ti) / allr;
    }
}

__global__ __launch_bounds__(256) void smoothap_row_reduce_kernel(
    const float* __restrict__ contrib, const float* __restrict__ targ,
    float* __restrict__ perRow) {
    __shared__ float an[256];
    __shared__ float ad[256];
    const int n   = blockIdx.x;
    const int tid = threadIdx.x;
    float sn = 0.0f, sd = 0.0f;
    for (int i = tid; i < M_DIM; i += 256) {
        sn += contrib[(size_t)n * M_DIM + i];
        float t = targ[(size_t)n * M_DIM + i];
        float p = (t > 0.5f) ? 1.0f : 0.0f;
        sd += p * t;
    }
    an[tid] = sn; ad[tid] = sd;
    __syncthreads();
    for (int s = 128; s > 0; s >>= 1) {
        if (tid < s) { an[tid] += an[tid + s]; ad[tid] += ad[tid + s]; }
        __syncthreads();
    }
    if (tid == 0) perRow[n] = an[0] / ad[0];
}

__global__ __launch_bounds__(32) void smoothap_finalize_kernel(
    const float* __restrict__ perRow, float* __restrict__ out, int N) {
    float v = 0.0f;
    for (int i = threadIdx.x; i < N; i += 32) v += perRow[i];
    for (int off = 16; off > 0; off >>= 1) v += __shfl_down(v, off, 32);
    if (threadIdx.x == 0) out[0] = 1.0f - v / (float)N;
}

extern "C" void kernel_launch(void* const* d_in, const int* in_sizes, int n_in,
                              void* d_out, int out_size, void* d_ws, size_t ws_size,
                              hipStream_t stream) {
    (void)n_in; (void)out_size; (void)ws_size;
    const float* sim  = (const float*)d_in[0];
    const float* targ = (const float*)d_in[1];
    float* out = (float*)d_out;

    const int total = in_sizes[0];       // N * M
    const int N = total / M_DIM;         // reference: 32

    float* contrib = (float*)d_ws;                       // N*M floats
    float* perRow  = contrib + (size_t)N * M_DIM;        // N floats

    dim3 g1(N * (M_DIM / TILE_I));
    smoothap_rank_kernel<<<g1, 32, 0, stream>>>(sim, targ, contrib);
    smoothap_row_reduce_kernel<<<N, 256, 0, stream>>>(contrib, targ, perRow);
    smoothap_finalize_kernel<<<1, 32, 0, stream>>>(perRow, out, N);
}